// Prediction_72241349919288
// MI455X (gfx1250) — compile-verified
//
#include <hip/hip_runtime.h>
#include <hip/hip_bf16.h>
#include <stdint.h>

#define BATCH   16
#define CCH     80
#define HH      128
#define WW      128
#define HW      (HH * WW)
#define CHW     (CCH * HW)
#define TOPK    100
#define NBINS   512
#define CAP     4096
#define RB      32              // rows per tile block
#define TPC     (HH / RB)       // 4 tiles per channel
#define NTILES  (BATCH * CCH * TPC)
#define SCALE_F 4.0f

// ---- workspace layout (bytes) ----
#define WS_HIST 0                                    // u32[BATCH][NBINS]
#define WS_THR  (WS_HIST + BATCH * NBINS * 4)        // u32[BATCH]
#define WS_CNT  (WS_THR + BATCH * 4)                 // u32[BATCH]
#define WS_CVAL (WS_CNT + BATCH * 4)                 // f32[BATCH][CAP]  (32896, 128B aligned)
#define WS_CIDX (WS_CVAL + BATCH * CAP * 4)          // u32[BATCH][CAP]

// ---------------- CDNA5 async global->LDS helpers ----------------
__device__ __forceinline__ void async_copy_b128(uint32_t lds_addr, const void* gaddr) {
  // ASYNCcnt-tracked DMA of 16B from global memory directly into LDS (gfx1250).
  asm volatile("global_load_async_to_lds_b128 %0, %1, off"
               :: "v"(lds_addr), "v"(gaddr) : "memory");
}
__device__ __forceinline__ void wait_async_zero() {
  asm volatile("s_wait_asynccnt 0x0" ::: "memory");
}
__device__ __forceinline__ uint32_t lds_addr_of(const void* shared_ptr) {
  // generic address of an LDS object: low 32 bits are the LDS byte address
  return (uint32_t)(uintptr_t)shared_ptr;
}

// Stage one (RB+2) x WW tile (with -inf halo rows) of channel data into LDS.
__device__ __forceinline__ void stage_tile(float* tile, const float* __restrict__ src,
                                           int y0, int tid) {
  const int NV4 = (RB + 2) * WW / 4;     // 1088 float4 chunks
  uint32_t base = lds_addr_of(tile);
  for (int v = tid; v < NV4; v += 256) {
    int r  = v >> 5;                     // WW/4 == 32 chunks per row
    int c4 = (v & 31) * 4;
    int gy = y0 - 1 + r;
    if (gy >= 0 && gy < HH) {
      async_copy_b128(base + (uint32_t)(v * 16), src + gy * WW + c4);
    } else {
      float4 ninf = make_float4(-__builtin_inff(), -__builtin_inff(),
                                -__builtin_inff(), -__builtin_inff());
      *(float4*)&tile[v * 4] = ninf;
    }
  }
  wait_async_zero();
  __syncthreads();
}

// NMS peak value at local element e (0..RB*WW-1); returns heatmap value if this
// pixel equals its 3x3 window max (reference: keep = (hmax == heatmap)), else 0.
__device__ __forceinline__ float peak_value(const float* tile, int e, int* ly_out, int* x_out) {
  int ly = e >> 7;                       // local row 0..RB-1
  int x  = e & (WW - 1);
  *ly_out = ly; *x_out = x;
  const float* r0 = &tile[(ly    ) * WW];
  const float* r1 = &tile[(ly + 1) * WW];
  const float* r2 = &tile[(ly + 2) * WW];
  float cen = r1[x];
  float m = cen;
  bool xm = (x > 0), xp = (x < WW - 1);
  if (xm) { m = fmaxf(m, r0[x-1]); m = fmaxf(m, r1[x-1]); m = fmaxf(m, r2[x-1]); }
  m = fmaxf(m, r0[x]); m = fmaxf(m, r2[x]);
  if (xp) { m = fmaxf(m, r0[x+1]); m = fmaxf(m, r1[x+1]); m = fmaxf(m, r2[x+1]); }
  return (cen >= m) ? cen : 0.0f;
}

// ---------------- kernels ----------------
__global__ __launch_bounds__(256) void k_zero(uint32_t* p, int n) {
  int i = blockIdx.x * 256 + threadIdx.x;
  if (i < n) p[i] = 0u;
}

__global__ __launch_bounds__(256) void k_hist(const float* __restrict__ heat,
                                              uint32_t* __restrict__ hist) {
  __shared__ __align__(16) float tile[(RB + 2) * WW];
  __shared__ uint32_t lh[NBINS];
  int tid = threadIdx.x;
  int blk = blockIdx.x;
  int b   = blk / (CCH * TPC);
  int rem = blk % (CCH * TPC);
  int c   = rem / TPC;
  int t   = rem % TPC;
  int y0  = t * RB;
  for (int i = tid; i < NBINS; i += 256) lh[i] = 0u;
  __syncthreads();
  stage_tile(tile, heat + ((size_t)b * CCH + c) * HW, y0, tid);
  for (int k = 0; k < (RB * WW) / 256; ++k) {
    int ly, x;
    float val = peak_value(tile, tid + k * 256, &ly, &x);
    if (val > 0.0f) {
      int bin = (int)(val * (float)NBINS);
      bin = bin < 0 ? 0 : (bin > NBINS - 1 ? NBINS - 1 : bin);
      atomicAdd(&lh[bin], 1u);
    }
  }
  __syncthreads();
  for (int i = tid; i < NBINS; i += 256)
    if (lh[i]) atomicAdd(&hist[b * NBINS + i], lh[i]);
}

__global__ __launch_bounds__(256) void k_thr(const uint32_t* __restrict__ hist,
                                             uint32_t* __restrict__ thr) {
  __shared__ uint32_t h[NBINS];
  int b = blockIdx.x;
  for (int i = threadIdx.x; i < NBINS; i += 256) h[i] = hist[b * NBINS + i];
  __syncthreads();
  if (threadIdx.x == 0) {
    uint32_t acc = 0; int bin = 0;
    for (int j = NBINS - 1; j >= 0; --j) { acc += h[j]; if (acc >= TOPK) { bin = j; break; } }
    thr[b] = (uint32_t)bin;
  }
}

__global__ __launch_bounds__(256) void k_collect(const float* __restrict__ heat,
                                                 const uint32_t* __restrict__ thr,
                                                 uint32_t* __restrict__ cnt,
                                                 float* __restrict__ cval,
                                                 uint32_t* __restrict__ cidx) {
  __shared__ __align__(16) float tile[(RB + 2) * WW];
  int tid = threadIdx.x;
  int blk = blockIdx.x;
  int b   = blk / (CCH * TPC);
  int rem = blk % (CCH * TPC);
  int c   = rem / TPC;
  int t   = rem % TPC;
  int y0  = t * RB;
  uint32_t bthr = thr[b];
  stage_tile(tile, heat + ((size_t)b * CCH + c) * HW, y0, tid);
  for (int k = 0; k < (RB * WW) / 256; ++k) {
    int ly, x;
    float val = peak_value(tile, tid + k * 256, &ly, &x);
    if (val > 0.0f) {
      int bin = (int)(val * (float)NBINS);
      bin = bin < 0 ? 0 : (bin > NBINS - 1 ? NBINS - 1 : bin);
      if ((uint32_t)bin >= bthr) {
        uint32_t pos = atomicAdd(&cnt[b], 1u);
        if (pos < CAP) {
          cval[b * CAP + pos] = val;
          cidx[b * CAP + pos] = (uint32_t)(c * HW + (y0 + ly) * WW + x);
        }
      }
    }
  }
}

__global__ __launch_bounds__(256) void k_final(const float* __restrict__ offp,
                                               const float* __restrict__ whp,
                                               const float* __restrict__ cval,
                                               const uint32_t* __restrict__ cidx,
                                               const uint32_t* __restrict__ cnt,
                                               float* __restrict__ out) {
  __shared__ float    v[CAP];
  __shared__ uint32_t ix[CAP];
  __shared__ float    rv[256];
  __shared__ uint32_t ri[256];
  __shared__ int      rs[256];
  __shared__ float    selV[TOPK];
  __shared__ uint32_t selI[TOPK];
  int b = blockIdx.x, tid = threadIdx.x;
  uint32_t n = cnt[b]; if (n > CAP) n = CAP;
  for (int i = tid; i < CAP; i += 256) {
    if (i < (int)n) { v[i] = cval[b * CAP + i]; ix[i] = cidx[b * CAP + i]; }
    else            { v[i] = 0.0f;              ix[i] = 0u; }
  }
  __syncthreads();
  for (int k = 0; k < TOPK; ++k) {
    float bv = -__builtin_inff(); uint32_t bi = 0xFFFFFFFFu; int bslot = -1;
    for (int i = tid; i < CAP; i += 256) {
      float val = v[i];
      if (val > bv || (val == bv && ix[i] < bi)) { bv = val; bi = ix[i]; bslot = i; }
    }
    rv[tid] = bv; ri[tid] = bi; rs[tid] = bslot;
    __syncthreads();
    for (int s = 128; s > 0; s >>= 1) {
      if (tid < s) {
        float ov = rv[tid + s]; uint32_t oi = ri[tid + s];
        if (ov > rv[tid] || (ov == rv[tid] && oi < ri[tid])) {
          rv[tid] = ov; ri[tid] = oi; rs[tid] = rs[tid + s];
        }
      }
      __syncthreads();
    }
    if (tid == 0) { selV[k] = rv[0]; selI[k] = ri[0]; if (rs[0] >= 0) v[rs[0]] = -__builtin_inff(); }
    __syncthreads();
  }
  if (tid < TOPK) {
    uint32_t idx = selI[tid];
    float score  = selV[tid];
    uint32_t cc  = idx / HW;
    uint32_t sp  = idx % HW;
    float ys = (float)(sp / WW), xs = (float)(sp % WW);
    const float* ob = offp + (size_t)b * 2 * HW;
    const float* wb = whp  + (size_t)b * 2 * HW;
    float ox = ob[sp], oy = ob[HW + sp];
    float bw = wb[sp], bh = wb[HW + sp];
    float cx = xs + ox, cy = ys + oy;
    float hw2 = bw * 0.5f, hh2 = bh * 0.5f;
    int ok = b * TOPK + tid;
    out[ok]                = (float)cc;   // ids
    out[BATCH*TOPK + ok]   = score;       // scores
    float* bb = out + 2 * BATCH * TOPK + (size_t)ok * 4;
    bb[0] = (cx - hw2) * SCALE_F;
    bb[1] = (cy - hh2) * SCALE_F;
    bb[2] = (cx + hw2) * SCALE_F;
    bb[3] = (cy + hh2) * SCALE_F;
  }
}

extern "C" void kernel_launch(void* const* d_in, const int* in_sizes, int n_in,
                              void* d_out, int out_size, void* d_ws, size_t ws_size,
                              hipStream_t stream) {
  const float* heat = (const float*)d_in[0];
  const float* offp = (const float*)d_in[1];
  const float* whp  = (const float*)d_in[2];
  float* out = (float*)d_out;
  (void)in_sizes; (void)n_in; (void)out_size; (void)ws_size;

  uint8_t* ws = (uint8_t*)d_ws;
  uint32_t* hist = (uint32_t*)(ws + WS_HIST);
  uint32_t* thr  = (uint32_t*)(ws + WS_THR);
  uint32_t* cnt  = (uint32_t*)(ws + WS_CNT);
  float*    cval = (float*)(ws + WS_CVAL);
  uint32_t* cidx = (uint32_t*)(ws + WS_CIDX);

  int nzero = BATCH * NBINS + 2 * BATCH;          // hist + thr + cnt (contiguous u32s)
  k_zero<<<(nzero + 255) / 256, 256, 0, stream>>>(hist, nzero);
  k_hist<<<NTILES, 256, 0, stream>>>(heat, hist);
  k_thr<<<BATCH, 256, 0, stream>>>(hist, thr);
  k_collect<<<NTILES, 256, 0, stream>>>(heat, thr, cnt, cval, cidx);
  k_final<<<BATCH, 256, 0, stream>>>(offp, whp, cval, cidx, cnt, out);
}